// Transformer_55516747268529
// MI455X (gfx1250) — compile-verified
//
#include <hip/hip_runtime.h>
#include <math.h>

#define NPTS 32768
#define DH   32
#define NH   8
#define NC   3
#define BSZ  128
#define NBK  256           // NPTS / BSZ
#define HD   256           // NH * DH
#define DR   35            // DH + 3
#define CHN  (NC*NH*NPTS)  // 786432

typedef __attribute__((ext_vector_type(2))) float v2f;
typedef __attribute__((ext_vector_type(8))) float v8f;

static __device__ __forceinline__ v8f wmma4(v2f a, v2f b, v8f c) {
  // D = A(16x4 f32) * B(4x16 f32) + C(16x16 f32)
  return __builtin_amdgcn_wmma_f32_16x16x4_f32(false, a, false, b, (short)0, c, false, false);
}

// monotone float <-> uint encoding for deterministic integer atomic min/max
static __device__ __forceinline__ unsigned enc_f(float f) {
  unsigned u = __float_as_uint(f);
  return (u & 0x80000000u) ? ~u : (u | 0x80000000u);
}
static __device__ __forceinline__ float dec_f(unsigned e) {
  unsigned b = (e & 0x80000000u) ? (e & 0x7FFFFFFFu) : ~e;
  return __uint_as_float(b);
}

// ---------------------------------------------------------------- init
__global__ void init_kernel(unsigned* umax, unsigned* umin) {
  int t = threadIdx.x;
  if (t < NC * NH) { umax[t] = 0u; umin[t] = 0xFFFFFFFFu; }
}

// ---------------------------------------------------------------- layernorm (D=32, thread per row)
__global__ __launch_bounds__(256) void ln_kernel(const float* __restrict__ x,
                                                 const float* __restrict__ w,
                                                 const float* __restrict__ b,
                                                 float* __restrict__ y) {
  int t = blockIdx.x * 256 + threadIdx.x;   // row
  float buf[32];
  float s = 0.f;
#pragma unroll
  for (int i = 0; i < 32; ++i) { buf[i] = x[(size_t)t * 32 + i]; s += buf[i]; }
  float mu = s * (1.f / 32.f);
  float var = 0.f;
#pragma unroll
  for (int i = 0; i < 32; ++i) { float d = buf[i] - mu; var += d * d; }
  var *= (1.f / 32.f);
  float rs = rsqrtf(var + 1e-5f);
#pragma unroll
  for (int i = 0; i < 32; ++i)
    y[(size_t)t * 32 + i] = (buf[i] - mu) * rs * w[i] + b[i];
}

// ---------------------------------------------------------------- QKV projection: [N,32] x [32,256] (x3)
__global__ __launch_bounds__(256) void qkv_kernel(const float* __restrict__ xn,
                                                  const float* __restrict__ wq,
                                                  const float* __restrict__ wk,
                                                  const float* __restrict__ wv,
                                                  float* __restrict__ q,
                                                  float* __restrict__ k,
                                                  float* __restrict__ v) {
  int lane = threadIdx.x & 31, wv_id = threadIdx.x >> 5;
  int mt = blockIdx.x;                    // 0..2047
  int nt = blockIdx.y * 8 + wv_id;        // 0..15
  int z  = blockIdx.z;
  const float* W  = (z == 0) ? wq : (z == 1 ? wk : wv);
  float*       out = (z == 0) ? q : (z == 1 ? k : v);
  int n = lane & 15, kh = lane >> 4;
  v8f acc = {};
#pragma unroll
  for (int kk = 0; kk < 32; kk += 4) {
    v2f a, bb;
    const float* ap = xn + (size_t)(mt * 16 + n) * 32 + kk + kh * 2;
    a.x = ap[0]; a.y = ap[1];
    bb.x = W[(kk + kh * 2 + 0) * HD + nt * 16 + n];
    bb.y = W[(kk + kh * 2 + 1) * HD + nt * 16 + n];
    acc = wmma4(a, bb, acc);
  }
#pragma unroll
  for (int r = 0; r < 8; ++r)
    out[(size_t)(mt * 16 + kh * 8 + r) * HD + nt * 16 + n] = acc[r];
}

// ---------------------------------------------------------------- RPE scale: sqrt(2*qw)[H,R]
__global__ void rpe_kernel(const float* __restrict__ wrpe, float* __restrict__ sqrtw) {
  int t = threadIdx.x;
  if (t < NH * 3) {
    int h = t / 3, r = t % 3;
    float qw = 0.f;
    for (int kk = 0; kk < 8; ++kk) {
      float s = 0.f;
      for (int d = 0; d < 32; ++d) s += wrpe[(h * 32 + d) * 24 + r * 8 + kk];
      qw += __expf(fminf(s, 50.f));
    }
    sqrtw[t] = sqrtf(2.f * qw);
  }
}

// ---------------------------------------------------------------- LSH hash values + global min/max
__global__ __launch_bounds__(256) void hash_kernel(const float* __restrict__ q,
                                                   const float* __restrict__ k,
                                                   const float* __restrict__ coords,
                                                   const float* __restrict__ sqrtw,
                                                   const float* __restrict__ alpha,
                                                   float* __restrict__ keys,
                                                   unsigned* __restrict__ umax,
                                                   unsigned* __restrict__ umin) {
  int t = blockIdx.x * 256 + threadIdx.x;   // 0 .. NH*NPTS-1
  int h = t >> 15, n = t & (NPTS - 1);
  float qa0 = 0, qa1 = 0, qa2 = 0, ka0 = 0, ka1 = 0, ka2 = 0;
  for (int d = 0; d < DR; ++d) {
    float qd, kd;
    if (d < 32) {
      qd = q[(size_t)n * HD + h * DH + d];
      kd = k[(size_t)n * HD + h * DH + d];
    } else {
      float cw = sqrtw[h * 3 + (d - 32)] * coords[n * 3 + (d - 32)];
      qd = cw; kd = cw;
    }
    const float* al = alpha + (h * DR + d) * NC;
    qa0 += qd * al[0]; qa1 += qd * al[1]; qa2 += qd * al[2];
    ka0 += kd * al[0]; ka1 += kd * al[1]; ka2 += kd * al[2];
  }
  float qv[3] = {qa0, qa1, qa2}, kv2[3] = {ka0, ka1, ka2};
  for (int c = 0; c < NC; ++c) {
    size_t idx = (size_t)c * NH * NPTS + (size_t)h * NPTS + n;
    keys[idx]       = qv[c];
    keys[CHN + idx] = kv2[c];
    unsigned eq = enc_f(qv[c]), ek = enc_f(kv2[c]);
    unsigned mx = eq > ek ? eq : ek;
    unsigned mn = eq < ek ? eq : ek;
    for (int m = 16; m; m >>= 1) {
      unsigned o1 = __shfl_xor(mx, m, 32); mx = mx > o1 ? mx : o1;
      unsigned o2 = __shfl_xor(mn, m, 32); mn = mn < o2 ? mn : o2;
    }
    if ((threadIdx.x & 31) == 0) {
      atomicMax(&umax[c * NH + h], mx);
      atomicMin(&umin[c * NH + h], mn);
    }
  }
}

// ---------------------------------------------------------------- add shifts + init index arrays
__global__ __launch_bounds__(256) void shift_kernel(float* __restrict__ keys,
                                                    int* __restrict__ idxs,
                                                    const int* __restrict__ cshift,
                                                    const unsigned* __restrict__ umax,
                                                    const unsigned* __restrict__ umin) {
  int t = blockIdx.x * 256 + threadIdx.x;   // 0 .. CHN-1, flat [c][h][n]
  int chIdx = t >> 15;
  float shift = dec_f(umax[chIdx]) - dec_f(umin[chIdx]);
  float sh = (float)cshift[t] * shift;
  keys[t]       += sh;
  keys[CHN + t] += sh;
  int n = t & (NPTS - 1);
  idxs[t]       = n;
  idxs[CHN + t] = n;
}

// ---------------------------------------------------------------- bitonic sort (48 segments of 32768)
__global__ __launch_bounds__(512) void bitonic_local_kernel(float* __restrict__ keys,
                                                            int* __restrict__ idxs) {
  __shared__ float sk[4096];
  __shared__ int   si[4096];
  int seg = blockIdx.y, chunk = blockIdx.x, tid = threadIdx.x;
  size_t base = (size_t)seg * NPTS + (size_t)chunk * 4096;
  int gb = chunk * 4096;
  for (int i = tid; i < 4096; i += 512) { sk[i] = keys[base + i]; si[i] = idxs[base + i]; }
  __syncthreads();
  for (int k = 2; k <= 4096; k <<= 1) {
    for (int j = k >> 1; j >= 1; j >>= 1) {
      for (int p = tid; p < 2048; p += 512) {
        int i  = ((p / j) * (j << 1)) + (p % j);
        int ip = i + j;
        bool up = (((gb + i) & k) == 0);
        float a = sk[i], c = sk[ip];
        bool sw = up ? (a > c) : (a < c);
        if (sw) { sk[i] = c; sk[ip] = a; int t0 = si[i]; si[i] = si[ip]; si[ip] = t0; }
      }
      __syncthreads();
    }
  }
  for (int i = tid; i < 4096; i += 512) { keys[base + i] = sk[i]; idxs[base + i] = si[i]; }
}

__global__ __launch_bounds__(256) void bitonic_global_kernel(float* __restrict__ keys,
                                                             int* __restrict__ idxs,
                                                             int k, int j) {
  int seg = blockIdx.y;
  int p = blockIdx.x * 256 + threadIdx.x;   // 0..16383
  int i  = ((p / j) * (j << 1)) + (p % j);
  int ip = i + j;
  bool up = ((i & k) == 0);
  size_t base = (size_t)seg * NPTS;
  float a = keys[base + i], c = keys[base + ip];
  bool sw = up ? (a > c) : (a < c);
  if (sw) {
    keys[base + i] = c; keys[base + ip] = a;
    int ta = idxs[base + i], tc = idxs[base + ip];
    idxs[base + i] = tc; idxs[base + ip] = ta;
  }
}

__global__ __launch_bounds__(512) void bitonic_local_final_kernel(float* __restrict__ keys,
                                                                  int* __restrict__ idxs,
                                                                  int k) {
  __shared__ float sk[4096];
  __shared__ int   si[4096];
  int seg = blockIdx.y, chunk = blockIdx.x, tid = threadIdx.x;
  size_t base = (size_t)seg * NPTS + (size_t)chunk * 4096;
  int gb = chunk * 4096;
  bool up = ((gb & k) == 0);           // k >= 8192, chunk aligned to 4096 -> uniform
  for (int i = tid; i < 4096; i += 512) { sk[i] = keys[base + i]; si[i] = idxs[base + i]; }
  __syncthreads();
  for (int j = 2048; j >= 1; j >>= 1) {
    for (int p = tid; p < 2048; p += 512) {
      int i  = ((p / j) * (j << 1)) + (p % j);
      int ip = i + j;
      float a = sk[i], c = sk[ip];
      bool sw = up ? (a > c) : (a < c);
      if (sw) { sk[i] = c; sk[ip] = a; int t0 = si[i]; si[i] = si[ip]; si[ip] = t0; }
    }
    __syncthreads();
  }
  for (int i = tid; i < 4096; i += 512) { keys[base + i] = sk[i]; idxs[base + i] = si[i]; }
}

// ---------------------------------------------------------------- inverse permutation of q_pos
__global__ __launch_bounds__(256) void qrev_kernel(const int* __restrict__ qpos,
                                                   int* __restrict__ qrev) {
  int t = blockIdx.x * 256 + threadIdx.x;   // 0 .. CHN-1
  int slot = t & (NPTS - 1);
  int base = t - slot;
  qrev[base + qpos[t]] = slot;
}

// ---------------------------------------------------------------- bucketed attention (WMMA f32)
__global__ __launch_bounds__(256) void attn_kernel(const float* __restrict__ q,
                                                   const float* __restrict__ kmat,
                                                   const float* __restrict__ vmat,
                                                   const float* __restrict__ coords,
                                                   const float* __restrict__ sqrtw,
                                                   const int* __restrict__ qpos,
                                                   const int* __restrict__ kpos,
                                                   float* __restrict__ so,
                                                   float* __restrict__ den) {
  __shared__ float Qs[BSZ * 36];
  __shared__ float Ks[BSZ * 36];
  __shared__ float Vs[BSZ * 33];
  __shared__ float qsq[BSZ];
  __shared__ float ksq[BSZ];
  __shared__ float Ts[8][16 * 17];

  int blk = blockIdx.x;          // ((c*8)+h)*256 + b
  int b   = blk & 255;
  int ch  = blk >> 8;            // c*8 + h
  int h   = ch & 7;
  int tid = threadIdx.x;
  const int* qp = qpos + (size_t)ch * NPTS + b * BSZ;
  const int* kp = kpos + (size_t)ch * NPTS + b * BSZ;

  if (tid < 128) {
    int row = tid;
    int nq = qp[row];
    float s = 0.f;
    for (int d = 0; d < 32; ++d) {
      float val = q[(size_t)nq * HD + h * DH + d];
      Qs[row * 36 + d] = val; s += val * val;
    }
    for (int r = 0; r < 3; ++r) {
      float val = sqrtw[h * 3 + r] * coords[nq * 3 + r];
      Qs[row * 36 + 32 + r] = val; s += val * val;
    }
    Qs[row * 36 + 35] = 0.f;
    qsq[row] = -0.5f * s;
  } else {
    int row = tid - 128;
    int nk = kp[row];
    float s = 0.f;
    for (int d = 0; d < 32; ++d) {
      float val = kmat[(size_t)nk * HD + h * DH + d];
      Ks[row * 36 + d] = val; s += val * val;
      Vs[row * 33 + d] = vmat[(size_t)nk * HD + h * DH + d];
    }
    for (int r = 0; r < 3; ++r) {
      float val = sqrtw[h * 3 + r] * coords[nk * 3 + r];
      Ks[row * 36 + 32 + r] = val; s += val * val;
    }
    Ks[row * 36 + 35] = 0.f;
    ksq[row] = -0.5f * s;
  }
  __syncthreads();

  int lane = tid & 31, w = tid >> 5;
  int i0 = w * 16;
  int nl = lane & 15, half = lane >> 4;
  float qs[8];
#pragma unroll
  for (int r = 0; r < 8; ++r) qs[r] = qsq[i0 + half * 8 + r];
  v8f acc2a = {}, acc2b = {};
  float rp[8] = {0, 0, 0, 0, 0, 0, 0, 0};
  float* tw = Ts[w];
  const float* Qrow = &Qs[(i0 + nl) * 36 + half * 2];

  for (int jt = 0; jt < 8; ++jt) {
    v8f acc = {};
    const float* Krow = &Ks[(jt * 16 + nl) * 36 + half * 2];
#pragma unroll
    for (int kk = 0; kk < 9; ++kk) {
      v2f a, bb;
      a.x = Qrow[kk * 4]; a.y = Qrow[kk * 4 + 1];
      bb.x = Krow[kk * 4]; bb.y = Krow[kk * 4 + 1];
      acc = wmma4(a, bb, acc);
    }
    float ks_n = ksq[jt * 16 + nl];
#pragma unroll
    for (int r = 0; r < 8; ++r) {
      float vqk = fminf(acc[r] + qs[r] + ks_n, 0.f);
      float e = __expf(vqk);
      rp[r] += e;
      tw[(half * 8 + r) * 17 + nl] = e;
    }
    __syncthreads();
    const float* Arow = &tw[nl * 17 + half * 2];
#pragma unroll
    for (int k2 = 0; k2 < 4; ++k2) {
      v2f a, b0, b1;
      a.x = Arow[k2 * 4]; a.y = Arow[k2 * 4 + 1];
      int kr = jt * 16 + k2 * 4 + half * 2;
      b0.x = Vs[(kr + 0) * 33 + nl];      b0.y = Vs[(kr + 1) * 33 + nl];
      b1.x = Vs[(kr + 0) * 33 + 16 + nl]; b1.y = Vs[(kr + 1) * 33 + 16 + nl];
      acc2a = wmma4(a, b0, acc2a);
      acc2b = wmma4(a, b1, acc2b);
    }
    __syncthreads();
  }
#pragma unroll
  for (int r = 0; r < 8; ++r) {
    float s = rp[r];
    for (int m = 1; m < 16; m <<= 1) s += __shfl_xor(s, m, 32);
    rp[r] = s;
  }
  size_t rowBase = (size_t)ch * NPTS + (size_t)b * BSZ + i0 + half * 8;
  if (nl == 0) {
#pragma unroll
    for (int r = 0; r < 8; ++r) den[rowBase + r] = rp[r] + 1e-20f;
  }
#pragma unroll
  for (int r = 0; r < 8; ++r) {
    size_t o = (rowBase + r) * 32;
    so[o + nl]      = acc2a[r];
    so[o + 16 + nl] = acc2b[r];
  }
}

// ---------------------------------------------------------------- inverse-gather combine over C hashes
__global__ __launch_bounds__(256) void combine_kernel(const float* __restrict__ so,
                                                      const float* __restrict__ den,
                                                      const int* __restrict__ qrev,
                                                      float* __restrict__ attn_cat) {
  int t = blockIdx.x * 256 + threadIdx.x;   // 0 .. N*H-1
  int n = t >> 3, h = t & 7;
  int slot[NC];
  float lg = 0.f;
#pragma unroll
  for (int c = 0; c < NC; ++c) {
    int ch = c * NH + h;
    slot[c] = qrev[(size_t)ch * NPTS + n];
    lg += den[(size_t)ch * NPTS + slot[c]];
  }
  float inv = 1.f / lg;
#pragma unroll
  for (int d4 = 0; d4 < 8; ++d4) {
    float ax = 0, ay = 0, az = 0, aw = 0;
#pragma unroll
    for (int c = 0; c < NC; ++c) {
      int ch = c * NH + h;
      const float4 vv = *(const float4*)(so + ((size_t)ch * NPTS + slot[c]) * 32 + d4 * 4);
      ax += vv.x; ay += vv.y; az += vv.z; aw += vv.w;
    }
    float4* o = (float4*)(attn_cat + (size_t)n * HD + h * DH + d4 * 4);
    *o = make_float4(ax * inv, ay * inv, az * inv, aw * inv);
  }
}

// ---------------------------------------------------------------- generic [N,K] x [K,32] GEMM (+bias, +res / silu)
__global__ __launch_bounds__(256) void gemm32_kernel(const float* __restrict__ A, int lda,
                                                     const float* __restrict__ W, int K,
                                                     const float* __restrict__ bias,
                                                     const float* __restrict__ res,
                                                     float* __restrict__ out, int mode) {
  int lane = threadIdx.x & 31, wv_id = threadIdx.x >> 5;
  int id = blockIdx.x * 8 + wv_id;
  int mt = id >> 1, nt = id & 1;
  int n = lane & 15, kh = lane >> 4;
  v8f acc = {};
  for (int kk = 0; kk < K; kk += 4) {
    v2f a, bb;
    const float* ap = A + (size_t)(mt * 16 + n) * lda + kk + kh * 2;
    a.x = ap[0]; a.y = ap[1];
    bb.x = W[(kk + kh * 2 + 0) * 32 + nt * 16 + n];
    bb.y = W[(kk + kh * 2 + 1) * 32 + nt * 16 + n];
    acc = wmma4(a, bb, acc);
  }
  int col = nt * 16 + n;
  float bc = bias[col];
#pragma unroll
  for (int r = 0; r < 8; ++r) {
    int row = mt * 16 + kh * 8 + r;
    float v = acc[r] + bc;
    if (mode == 1) v = v / (1.f + __expf(-v));          // silu
    else if (res) v += res[(size_t)row * 32 + col];     // residual
    out[(size_t)row * 32 + col] = v;
  }
}

// ================================================================ host
extern "C" void kernel_launch(void* const* d_in, const int* in_sizes, int n_in,
                              void* d_out, int out_size, void* d_ws, size_t ws_size,
                              hipStream_t stream) {
  (void)in_sizes; (void)n_in; (void)out_size; (void)ws_size;
  const float* x      = (const float*)d_in[0];
  const float* coords = (const float*)d_in[1];
  const int*   cshift = (const int*)d_in[2];
  const float* n1w    = (const float*)d_in[3];
  const float* n1b    = (const float*)d_in[4];
  const float* wq     = (const float*)d_in[5];
  const float* wk     = (const float*)d_in[6];
  const float* wv     = (const float*)d_in[7];
  const float* wrpe   = (const float*)d_in[8];
  const float* alpha  = (const float*)d_in[9];
  const float* outw   = (const float*)d_in[10];
  const float* outb   = (const float*)d_in[11];
  const float* n2w    = (const float*)d_in[12];
  const float* n2b    = (const float*)d_in[13];
  const float* ff1w   = (const float*)d_in[14];
  const float* ff1b   = (const float*)d_in[15];
  const float* ff2w   = (const float*)d_in[16];
  const float* ff2b   = (const float*)d_in[17];

  float* WS = (float*)d_ws;
  size_t off = 0;
  float*    xn    = WS + off; off += (size_t)NPTS * DH;   // later reused as x1
  float*    qb    = WS + off; off += (size_t)NPTS * HD;
  float*    kb    = WS + off; off += (size_t)NPTS * HD;   // later reused as attn_cat
  float*    vb    = WS + off; off += (size_t)NPTS * HD;   // later reused as xn2 / ffh
  float*    keys  = WS + off; off += (size_t)2 * CHN;
  int*      idxs  = (int*)(WS + off); off += (size_t)2 * CHN;
  int*      qrev  = (int*)(WS + off); off += (size_t)CHN;
  float*    sqw   = WS + off; off += 64;
  unsigned* umax  = (unsigned*)(WS + off); off += 32;
  unsigned* umin  = (unsigned*)(WS + off); off += 32;
  float*    den   = WS + off; off += (size_t)CHN;
  float*    soA   = WS + off; off += (size_t)CHN * 32;

  init_kernel<<<1, 64, 0, stream>>>(umax, umin);
  ln_kernel<<<NPTS / 256, 256, 0, stream>>>(x, n1w, n1b, xn);
  qkv_kernel<<<dim3(NPTS / 16, 2, 3), 256, 0, stream>>>(xn, wq, wk, wv, qb, kb, vb);
  rpe_kernel<<<1, 32, 0, stream>>>(wrpe, sqw);
  hash_kernel<<<(NH * NPTS) / 256, 256, 0, stream>>>(qb, kb, coords, sqw, alpha, keys, umax, umin);
  shift_kernel<<<CHN / 256, 256, 0, stream>>>(keys, idxs, cshift, umax, umin);

  // segmented bitonic argsort of 48 segments (q-keys and k-keys)
  bitonic_local_kernel<<<dim3(8, 48), 512, 0, stream>>>(keys, idxs);
  for (int k = 8192; k <= 32768; k <<= 1) {
    for (int j = k >> 1; j >= 4096; j >>= 1)
      bitonic_global_kernel<<<dim3(64, 48), 256, 0, stream>>>(keys, idxs, k, j);
    bitonic_local_final_kernel<<<dim3(8, 48), 512, 0, stream>>>(keys, idxs, k);
  }

  qrev_kernel<<<CHN / 256, 256, 0, stream>>>(idxs, qrev);
  attn_kernel<<<NC * NH * NBK, 256, 0, stream>>>(qb, kb, vb, coords, sqw,
                                                 idxs, idxs + CHN, soA, den);
  combine_kernel<<<(NPTS * NH) / 256, 256, 0, stream>>>(soA, den, qrev, kb);

  // out projection + residual -> x1 (in xn)
  gemm32_kernel<<<512, 256, 0, stream>>>(kb, HD, outw, HD, outb, x, xn, 0);
  // LN2 -> xn2 (in vb)
  ln_kernel<<<NPTS / 256, 256, 0, stream>>>(xn, n2w, n2b, vb);
  // FF1 + silu -> ffh (vb + N*32)
  gemm32_kernel<<<512, 256, 0, stream>>>(vb, DH, ff1w, DH, ff1b, nullptr, vb + (size_t)NPTS * DH, 1);
  // FF2 + residual(x1) -> output
  gemm32_kernel<<<512, 256, 0, stream>>>(vb + (size_t)NPTS * DH, DH, ff2w, DH, ff2b, xn, (float*)d_out, 0);
}